// SAVSS_2D_DirConv_45681272160361
// MI455X (gfx1250) — compile-verified
//
#include <hip/hip_runtime.h>
#include <math.h>

// ---------------- problem constants (fixed by setup_inputs) ----------------
#define B0   2
#define H0   64
#define W0   64
#define L0   4096          // H0*W0
#define C0   96
#define E0   192
#define N0   16
#define R0   6
#define M0   8192          // B0*L0
#define K4E  768           // 4*E0
#define XD   38            // R0 + 2*N0

typedef float v2f __attribute__((ext_vector_type(2)));
typedef float v8f __attribute__((ext_vector_type(8)));

__device__ __forceinline__ float silu_f(float x) {
  return x / (1.0f + __expf(-x));
}
__device__ __forceinline__ float softplus_f(float x) {
  return (x > 20.0f) ? x : log1pf(__expf(x));
}

// ---------------------------------------------------------------------------
// GEMM: out[m,n] = sum_k X[m*ldx+k] * W[n*ldw+k] (+ bias[n])
// Requirements (all satisfied by every call here): M % 16 == 0, K % 16 == 0.
// One wave32 computes a 16x32 output tile (two 16x16 WMMA accumulators that
// share the A fragment). K unrolled by 16: 12 unconditional b64 loads issued
// ahead of 8 dependent V_WMMA_F32_16X16X4_F32 ops. Ragged N handled by
// clamping the B row index (harmless: a B column only feeds its own output
// column, which the store guard discards).
// ---------------------------------------------------------------------------
__global__ void gemm_xwt_wmma(const float* __restrict__ X,
                              const float* __restrict__ Wt,
                              const float* __restrict__ bias,
                              float* __restrict__ out,
                              int M, int N, int K, int ldx, int ldw, int ldo) {
  int tilesN = (N + 31) >> 5;
  int tilesM = M >> 4;
  int tile = blockIdx.x * (blockDim.x >> 5) + (threadIdx.x >> 5);
  if (tile >= tilesM * tilesN) return;
  int tm = tile / tilesN;
  int tn = tile % tilesN;

  int lane = threadIdx.x & 31;
  int half = lane >> 4;      // 0 or 1 -> K pair {0,1} / {2,3} within each k4
  int lm   = lane & 15;

  int arow = tm * 16 + lm;
  int bc0  = tn * 32 + lm;
  int bc1  = tn * 32 + 16 + lm;
  int bc0c = bc0 < N ? bc0 : N - 1;   // clamp: OOB columns never stored
  int bc1c = bc1 < N ? bc1 : N - 1;

  const float* __restrict__ xr  = X  + (size_t)arow * ldx + 2 * half;
  const float* __restrict__ wr0 = Wt + (size_t)bc0c * ldw + 2 * half;
  const float* __restrict__ wr1 = Wt + (size_t)bc1c * ldw + 2 * half;

  v8f acc0 = {0.f, 0.f, 0.f, 0.f, 0.f, 0.f, 0.f, 0.f};
  v8f acc1 = {0.f, 0.f, 0.f, 0.f, 0.f, 0.f, 0.f, 0.f};

  for (int k = 0; k < K; k += 16) {
    // issue all loads for this K chunk first (b64 each)
    v2f a0 = *(const v2f*)(xr + k);
    v2f a1 = *(const v2f*)(xr + k + 4);
    v2f a2 = *(const v2f*)(xr + k + 8);
    v2f a3 = *(const v2f*)(xr + k + 12);
    v2f p0 = *(const v2f*)(wr0 + k);
    v2f p1 = *(const v2f*)(wr0 + k + 4);
    v2f p2 = *(const v2f*)(wr0 + k + 8);
    v2f p3 = *(const v2f*)(wr0 + k + 12);
    v2f q0 = *(const v2f*)(wr1 + k);
    v2f q1 = *(const v2f*)(wr1 + k + 4);
    v2f q2 = *(const v2f*)(wr1 + k + 8);
    v2f q3 = *(const v2f*)(wr1 + k + 12);

    acc0 = __builtin_amdgcn_wmma_f32_16x16x4_f32(false, a0, false, p0, (short)0, acc0, false, false);
    acc1 = __builtin_amdgcn_wmma_f32_16x16x4_f32(false, a0, false, q0, (short)0, acc1, false, false);
    acc0 = __builtin_amdgcn_wmma_f32_16x16x4_f32(false, a1, false, p1, (short)0, acc0, false, false);
    acc1 = __builtin_amdgcn_wmma_f32_16x16x4_f32(false, a1, false, q1, (short)0, acc1, false, false);
    acc0 = __builtin_amdgcn_wmma_f32_16x16x4_f32(false, a2, false, p2, (short)0, acc0, false, false);
    acc1 = __builtin_amdgcn_wmma_f32_16x16x4_f32(false, a2, false, q2, (short)0, acc1, false, false);
    acc0 = __builtin_amdgcn_wmma_f32_16x16x4_f32(false, a3, false, p3, (short)0, acc0, false, false);
    acc1 = __builtin_amdgcn_wmma_f32_16x16x4_f32(false, a3, false, q3, (short)0, acc1, false, false);
  }

  // D layout: VGPR r -> row tm*16 + half*8 + r, col = tile col base + lm
  int orow0 = tm * 16 + half * 8;
  if (bc0 < N) {
    float bv = bias ? bias[bc0] : 0.0f;
#pragma unroll
    for (int r = 0; r < 8; ++r)
      out[(size_t)(orow0 + r) * ldo + bc0] = acc0[r] + bv;
  }
  if (bc1 < N) {
    float bv = bias ? bias[bc1] : 0.0f;
#pragma unroll
    for (int r = 0; r < 8; ++r)
      out[(size_t)(orow0 + r) * ldo + bc1] = acc1[r] + bv;
  }
}

// ---------------------------------------------------------------------------
// Build the four scan orders on device (replicates Python sass(); H even path).
// ---------------------------------------------------------------------------
__global__ void build_orders_kernel(int* __restrict__ orders) {
  int d = threadIdx.x;
  if (d >= 4) return;
  int* o = orders + d * L0;
  const int H = H0, W = W0;
  if (d == 0) {               // bottom snake (H even: start bottom-left, right)
    int i = H - 1, j = 0, jd = 1, pos = 0;
    while (i > -1) {
      o[pos++] = i * W + j;
      if (jd == 1) { if (j < W - 1) j++; else { i--; jd = 0; } }
      else         { if (j > 0)     j--; else { i--; jd = 1; } }
    }
  } else if (d == 1) {        // column snake from (0,0) going down
    int i = 0, j = 0, dd = 1, pos = 0;
    while (j < W) {
      o[pos++] = i * W + j;
      if (dd == 1) { if (i < H - 1) i++; else { j++; dd = 0; } }
      else         { if (i > 0)     i--; else { j++; dd = 1; } }
    }
  } else if (d == 2) {        // diagonals
    int pos = 0;
    for (int diag = 0; diag < H + W - 1; ++diag) {
      int imax = (diag + 1 < H) ? (diag + 1) : H;
      for (int i = 0; i < imax; ++i) {
        int j = diag - i;
        if (j < W) o[pos++] = i * W + j;
      }
    }
  } else {                    // anti-diagonals
    int pos = 0;
    for (int diag = 0; diag < H + W - 1; ++diag) {
      int imax = (diag + 1 < H) ? (diag + 1) : H;
      for (int i = 0; i < imax; ++i) {
        int j = diag - i;
        if (j < W) o[pos++] = i * W + (W - j - 1);
      }
    }
  }
}

__global__ void zero_kernel(float* __restrict__ p, int n) {
  int t = blockIdx.x * blockDim.x + threadIdx.x;
  if (t < n) p[t] = 0.0f;
}

// ---------------------------------------------------------------------------
// Directional convs (grouped, 2 out-chan per in-chan) + SiLU, written in
// [M, 4E] "sequence" layout so the fuse 1x1 becomes a WMMA GEMM.
// ---------------------------------------------------------------------------
__global__ void conv_kernel(const float* __restrict__ x,     // [B,L,C]
                            const float* __restrict__ wh,    // [E,7]
                            const float* __restrict__ wv,    // [E,7]
                            const float* __restrict__ wd1,   // [E,49]
                            const float* __restrict__ wd2,   // [E,49]
                            float* __restrict__ cat) {       // [M,4E]
  int t = blockIdx.x * blockDim.x + threadIdx.x;
  if (t >= M0 * E0) return;
  int e = t % E0;
  int m = t / E0;                 // b*L + h*W + w
  int hw = m % L0;
  int b  = m / L0;
  int h  = hw / W0;
  int w  = hw % W0;
  int c  = e >> 1;                // group mapping: 2 outputs per input channel

  const float* __restrict__ xb = x + (size_t)b * L0 * C0;
#define XPIX(hh, ww) xb[((hh) * W0 + (ww)) * C0 + c]

  // horizontal 1x7 (pad W by 3)
  float sh = 0.0f;
#pragma unroll
  for (int k = 0; k < 7; ++k) {
    int ww = w + k - 3;
    if (ww >= 0 && ww < W0) sh += wh[e * 7 + k] * XPIX(h, ww);
  }
  // vertical 7x1 (pad H by 3)
  float sv = 0.0f;
#pragma unroll
  for (int k = 0; k < 7; ++k) {
    int hh = h + k - 3;
    if (hh >= 0 && hh < H0) sv += wv[e * 7 + k] * XPIX(hh, w);
  }
  // two full 7x7
  float s1 = 0.0f, s2 = 0.0f;
  for (int kh = 0; kh < 7; ++kh) {
    int hh = h + kh - 3;
    if (hh < 0 || hh >= H0) continue;
    for (int kw = 0; kw < 7; ++kw) {
      int ww = w + kw - 3;
      if (ww < 0 || ww >= W0) continue;
      float xv2 = XPIX(hh, ww);
      s1 += wd1[e * 49 + kh * 7 + kw] * xv2;
      s2 += wd2[e * 49 + kh * 7 + kw] * xv2;
    }
  }
#undef XPIX
  float* row = cat + (size_t)m * K4E;
  row[e]            = silu_f(sh);
  row[E0 + e]       = silu_f(sv);
  row[2 * E0 + e]   = silu_f(s1);
  row[3 * E0 + e]   = silu_f(s2);
}

// ---------------------------------------------------------------------------
// dt = xdbl[:, :R] @ dt_proj_w.T + dt_proj_b   -> [M, E] (K=6, plain VALU)
// ---------------------------------------------------------------------------
__global__ void dt_kernel(const float* __restrict__ xdbl,
                          const float* __restrict__ dtw,   // [E,R]
                          const float* __restrict__ dtb,   // [E]
                          float* __restrict__ dt) {
  int t = blockIdx.x * blockDim.x + threadIdx.x;
  if (t >= M0 * E0) return;
  int e = t % E0;
  int m = t / E0;
  const float* xr = xdbl + (size_t)m * XD;
  float s = dtb[e];
#pragma unroll
  for (int r = 0; r < R0; ++r) s += xr[r] * dtw[e * R0 + r];
  dt[t] = s;
}

// ---------------------------------------------------------------------------
// Selective scan. One wave32 per 2 (b,e) pairs for one direction; lane = state
// index n. Faithful to reference: u is permuted by order o, delta/B/C are in
// raster order, dt_proj_b is added a second time inside softplus.
// y scattered with atomicAdd to ys[b, o[p], e].
// ---------------------------------------------------------------------------
__global__ void scan_kernel(const float* __restrict__ xz,     // [M,2E]; xp = cols [0,E)
                            const float* __restrict__ dt,     // [M,E]
                            const float* __restrict__ xdbl,   // [M,38]
                            const float* __restrict__ A_log,  // [E,N]
                            const float* __restrict__ Dp,     // [E]
                            const float* __restrict__ dtb,    // [E]
                            const float* __restrict__ dirB,   // [5,N]
                            const int*   __restrict__ orders, // [4,L]
                            float* __restrict__ ys) {         // [M,E]
  const int pairsPerDir = (B0 * E0) / 2;      // 192
  int blk  = blockIdx.x;
  int d    = blk / pairsPerDir;
  int pair = blk % pairsPerDir;
  int lane = threadIdx.x;
  int sub  = lane >> 4;                       // which (b,e) of the two
  int n    = lane & 15;
  int be   = pair * 2 + sub;
  int b    = be / E0;
  int e    = be % E0;

  const int* __restrict__ o = orders + d * L0;
  float Aen   = -__expf(A_log[e * N0 + n]);
  float Bdir  = dirB[d * N0 + n];
  float Dv    = Dp[e];
  float dbias = dtb[e];
  float h     = 0.0f;

  const size_t baseM = (size_t)b * L0;
  for (int p = 0; p < L0; ++p) {
    size_t m = baseM + p;
    float delta = softplus_f(dt[m * E0 + e] + dbias);
    int   idx   = o[p];
    float u     = xz[(baseM + idx) * (2 * E0) + e];
    const float* xr = xdbl + m * XD;
    float Bm = xr[R0 + n] + Bdir;
    float Cm = xr[R0 + N0 + n];
    float dA = __expf(delta * Aen);
    h = dA * h + delta * Bm * u;
    float yc = h * Cm;
#pragma unroll
    for (int off = 8; off; off >>= 1) yc += __shfl_xor(yc, off, 32);
    if (n == 0) atomicAdd(&ys[(baseM + idx) * E0 + e], yc + Dv * u);
  }
}

// ---------------------------------------------------------------------------
// Local clustering branch. One block (64 threads) per 8x8 window (128 blocks).
// Produces clout in [B, L, C] layout (post p_w projection).
// ---------------------------------------------------------------------------
__global__ void cluster_kernel(const float* __restrict__ seq,   // [B,L,C] (x_conv)
                               const float* __restrict__ f_w, const float* __restrict__ f_b,
                               const float* __restrict__ v_w, const float* __restrict__ v_b,
                               const float* __restrict__ p_w, const float* __restrict__ p_b,
                               const float* __restrict__ alpha_p,
                               const float* __restrict__ beta_p,
                               float* __restrict__ clout) {     // [B,L,C]
  __shared__ float fS[6][64], vS[6][64];
  __shared__ float cenN[6][25], vcS[6][25], cenS[6][25];
  __shared__ float simS[25][64];
  __shared__ float aggS[25][6];
  __shared__ int   amax[64];
  __shared__ float smax[64];

  int bb   = blockIdx.x;
  int b    = bb >> 6;
  int a    = (bb >> 3) & 7;       // coarse H
  int cblk = bb & 7;              // coarse W
  int p    = threadIdx.x;         // 0..63 : (wi, hi)
  int wi   = p >> 3, hi = p & 7;
  int sp   = (a * 8 + wi) * W0 + cblk * 8 + hi;
  const float* __restrict__ xr = seq + ((size_t)b * L0 + sp) * C0;

  for (int c = 0; c < 6; ++c) {
    float fa = f_b[c], va = v_b[c];
    for (int ci = 0; ci < 48; ++ci) {
      fa += f_w[c * 48 + ci] * xr[ci];
      va += v_w[c * 48 + ci] * xr[48 + ci];
    }
    fS[c][p] = fa;
    vS[c][p] = va;
  }
  __syncthreads();

  if (p < 25) {                    // adaptive-avg-pool 8 -> 5 centers
    int i = p / 5, j = p % 5;
    int s0 = (i * 8) / 5, e0i = ((i + 1) * 8 + 4) / 5;
    int s1 = (j * 8) / 5, e1i = ((j + 1) * 8 + 4) / 5;
    float wgt = 1.0f / (float)((e0i - s0) * (e1i - s1));
    float nn = 0.0f;
    for (int c = 0; c < 6; ++c) {
      float acf = 0.0f, acv = 0.0f;
      for (int ww = s0; ww < e0i; ++ww)
        for (int hh = s1; hh < e1i; ++hh) {
          acf += fS[c][ww * 8 + hh];
          acv += vS[c][ww * 8 + hh];
        }
      cenS[c][p] = acf * wgt;
      vcS[c][p]  = acv * wgt;
      nn += acf * wgt * acf * wgt;
    }
    nn = fmaxf(sqrtf(nn), 1e-12f);
    for (int c = 0; c < 6; ++c) cenN[c][p] = cenS[c][p] / nn;
  }
  __syncthreads();

  // normalized features + similarity + per-pixel argmax over 25 centers
  float nn = 0.0f;
  for (int c = 0; c < 6; ++c) nn += fS[c][p] * fS[c][p];
  nn = fmaxf(sqrtf(nn), 1e-12f);
  float fn[6];
  for (int c = 0; c < 6; ++c) fn[c] = fS[c][p] / nn;
  float alpha = alpha_p[0], beta = beta_p[0];
  float best = -1e30f;
  int   bk   = 0;
  for (int k = 0; k < 25; ++k) {
    float dot = 0.0f;
    for (int c = 0; c < 6; ++c) dot += cenN[c][k] * fn[c];
    float s = 1.0f / (1.0f + __expf(-(beta + alpha * dot)));
    simS[k][p] = s;
    if (s > best) { best = s; bk = k; }
  }
  amax[p] = bk;
  smax[p] = best;
  __syncthreads();

  if (p < 25) {                    // aggregate (masked sim)
    float av[6];
    for (int c = 0; c < 6; ++c) av[c] = vcS[c][p];
    float ssum = 0.0f;
    for (int q = 0; q < 64; ++q) {
      if (amax[q] == p) {
        float s = simS[p][q];
        ssum += s;
        for (int c = 0; c < 6; ++c) av[c] += s * vS[c][q];
      }
    }
    float inv = 1.0f / (ssum + 1.0f);
    for (int c = 0; c < 6; ++c) aggS[p][c] = av[c] * inv;
  }
  __syncthreads();

  // redistribute + 1x1 projection to C channels
  float oc[6];
  int k0 = amax[p];
  for (int c = 0; c < 6; ++c) oc[c] = aggS[k0][c] * smax[p];
  float* orow = clout + ((size_t)b * L0 + sp) * C0;
  for (int o = 0; o < C0; ++o) {
    float s = p_b[o];
    for (int c = 0; c < 6; ++c) s += p_w[o * 6 + c] * oc[c];
    orow[o] = s;
  }
}

// ---------------------------------------------------------------------------
// Per-row LayerNorm x3 + branch softmax mix. One wave32 per row (6 elems/lane).
// ---------------------------------------------------------------------------
__global__ void mix_ln_kernel(const float* __restrict__ ys,
                              const float* __restrict__ xz,   // zp = cols [E,2E)
                              const float* __restrict__ cl,
                              const float* __restrict__ ln_w,
                              const float* __restrict__ ln_b,
                              const float* __restrict__ bc,   // [3,E]
                              float* __restrict__ ymix) {
  int wave = (blockIdx.x * blockDim.x + threadIdx.x) >> 5;
  if (wave >= M0) return;
  int lane = threadIdx.x & 31;
  const float* yr = ys + (size_t)wave * E0;
  const float* zr = xz + (size_t)wave * (2 * E0) + E0;
  const float* cr = cl + (size_t)wave * E0;

  float av[6], zv[6], cv[6];
  float s1 = 0, s2 = 0, s3 = 0, q1 = 0, q2 = 0, q3 = 0;
#pragma unroll
  for (int i = 0; i < 6; ++i) {
    int e = lane * 6 + i;
    av[i] = yr[e]; zv[i] = zr[e]; cv[i] = cr[e];
    s1 += av[i]; q1 += av[i] * av[i];
    s2 += zv[i]; q2 += zv[i] * zv[i];
    s3 += cv[i]; q3 += cv[i] * cv[i];
  }
#pragma unroll
  for (int off = 16; off; off >>= 1) {
    s1 += __shfl_xor(s1, off, 32); q1 += __shfl_xor(q1, off, 32);
    s2 += __shfl_xor(s2, off, 32); q2 += __shfl_xor(q2, off, 32);
    s3 += __shfl_xor(s3, off, 32); q3 += __shfl_xor(q3, off, 32);
  }
  const float invE = 1.0f / (float)E0;
  float m1 = s1 * invE, m2 = s2 * invE, m3 = s3 * invE;
  float r1 = rsqrtf(q1 * invE - m1 * m1 + 1e-5f);
  float r2 = rsqrtf(q2 * invE - m2 * m2 + 1e-5f);
  float r3 = rsqrtf(q3 * invE - m3 * m3 + 1e-5f);

  float* orow = ymix + (size_t)wave * E0;
#pragma unroll
  for (int i = 0; i < 6; ++i) {
    int e = lane * 6 + i;
    float b0 = bc[e], b1 = bc[E0 + e], b2 = bc[2 * E0 + e];
    float mx = fmaxf(b0, fmaxf(b1, b2));
    float e0 = __expf(b0 - mx), e1 = __expf(b1 - mx), e2 = __expf(b2 - mx);
    float inv = 1.0f / (e0 + e1 + e2);
    float lw = ln_w[e], lb = ln_b[e];
    float l1 = (av[i] - m1) * r1 * lw + lb;
    float l2 = (zv[i] - m2) * r2 * lw + lb;
    float l3 = (cv[i] - m3) * r3 * lw + lb;
    orow[e] = (e0 * l1 + e1 * l2 + e2 * l3) * inv;
  }
}

// ---------------------------------------------------------------------------
extern "C" void kernel_launch(void* const* d_in, const int* in_sizes, int n_in,
                              void* d_out, int out_size, void* d_ws, size_t ws_size,
                              hipStream_t stream) {
  (void)in_sizes; (void)n_in; (void)out_size; (void)ws_size;
  const float* x        = (const float*)d_in[0];
  // d_in[1], d_in[2] = H, W (ints) — compile-time constants here
  const float* conv_h_w = (const float*)d_in[3];
  const float* conv_v_w = (const float*)d_in[4];
  const float* conv_d1w = (const float*)d_in[5];
  const float* conv_d2w = (const float*)d_in[6];
  const float* fuse_w   = (const float*)d_in[7];
  const float* in_projw = (const float*)d_in[8];
  const float* x_projw  = (const float*)d_in[9];
  const float* dt_projw = (const float*)d_in[10];
  const float* dt_projb = (const float*)d_in[11];
  const float* A_log    = (const float*)d_in[12];
  const float* Dp       = (const float*)d_in[13];
  const float* dirB     = (const float*)d_in[14];
  const float* out_projw= (const float*)d_in[15];
  const float* sim_a    = (const float*)d_in[16];
  const float* sim_b    = (const float*)d_in[17];
  const float* f_w      = (const float*)d_in[18];
  const float* f_b      = (const float*)d_in[19];
  const float* v_w      = (const float*)d_in[20];
  const float* v_b      = (const float*)d_in[21];
  const float* p_w      = (const float*)d_in[22];
  const float* p_b      = (const float*)d_in[23];
  const float* cl_projw = (const float*)d_in[24];
  const float* cl_projb = (const float*)d_in[25];
  const float* branch   = (const float*)d_in[26];
  const float* ln_w     = (const float*)d_in[27];
  const float* ln_b     = (const float*)d_in[28];
  float* out = (float*)d_out;

  // workspace layout (floats)
  float* ws = (float*)d_ws;
  float* cat    = ws;                                 // [M,768]
  float* seq    = cat   + (size_t)M0 * K4E;           // [M,96]
  float* xz     = seq   + (size_t)M0 * C0;            // [M,384]
  float* xdbl   = xz    + (size_t)M0 * 2 * E0;        // [M,38]
  float* dt     = xdbl  + (size_t)M0 * XD;            // [M,192]
  float* ysb    = dt    + (size_t)M0 * E0;            // [M,192]
  float* clout  = ysb   + (size_t)M0 * E0;            // [M,96]
  float* cl     = clout + (size_t)M0 * C0;            // [M,192]
  float* ymix   = cl    + (size_t)M0 * E0;            // [M,192]
  int*   orders = (int*)(ymix + (size_t)M0 * E0);     // [4,L]

  // 1. scan orders
  build_orders_kernel<<<1, 4, 0, stream>>>(orders);
  // 2. zero ys accumulator
  zero_kernel<<<(M0 * E0 + 255) / 256, 256, 0, stream>>>(ysb, M0 * E0);
  // 3. directional convs + SiLU -> concat [M,768]
  conv_kernel<<<(M0 * E0 + 255) / 256, 256, 0, stream>>>(
      x, conv_h_w, conv_v_w, conv_d1w, conv_d2w, cat);

  auto gemm = [&](const float* X, const float* Wm, const float* bias, float* O,
                  int M, int N, int K, int ldx, int ldw, int ldo) {
    int tiles  = (M / 16) * ((N + 31) / 32);           // 16x32 tile per wave
    int blocks = (tiles + 3) / 4;                      // 4 waves / block
    gemm_xwt_wmma<<<blocks, 128, 0, stream>>>(X, Wm, bias, O, M, N, K, ldx, ldw, ldo);
  };

  // 4. fuse 1x1: seq = cat @ fuse_w.T            [M,768] x [96,768] -> [M,96]
  gemm(cat, fuse_w, nullptr, seq, M0, C0, K4E, K4E, K4E, C0);
  // 5. in_proj: xz = seq @ in_proj_w.T           [M,96] x [384,96] -> [M,384]
  gemm(seq, in_projw, nullptr, xz, M0, 2 * E0, C0, C0, C0, 2 * E0);
  // 6. x_proj: xdbl = xp @ x_proj_w.T            xp = view of xz (ldx=384)
  gemm(xz, x_projw, nullptr, xdbl, M0, XD, E0, 2 * E0, E0, XD);
  // 7. dt = xdbl[:, :R] @ dt_proj_w.T + b
  dt_kernel<<<(M0 * E0 + 255) / 256, 256, 0, stream>>>(xdbl, dt_projw, dt_projb, dt);
  // 8. selective scan, 4 directions concurrently (atomic scatter into ys)
  scan_kernel<<<4 * (B0 * E0 / 2), 32, 0, stream>>>(
      xz, dt, xdbl, A_log, Dp, dt_projb, dirB, orders, ysb);
  // 9. local clustering -> clout [M,96]
  cluster_kernel<<<B0 * 64, 64, 0, stream>>>(
      seq, f_w, f_b, v_w, v_b, p_w, p_b, sim_a, sim_b, clout);
  // 10. cluster_proj: cl = clout @ w.T + b       [M,96] x [192,96] -> [M,192]
  gemm(clout, cl_projw, cl_projb, cl, M0, E0, C0, C0, C0, E0);
  // 11. triple LayerNorm + branch-softmax mix -> ymix [M,192]
  mix_ln_kernel<<<(M0 * 32 + 255) / 256, 256, 0, stream>>>(
      ysb, xz, cl, ln_w, ln_b, branch, ymix);
  // 12. out_proj: out = ymix @ out_proj_w.T      [M,192] x [96,192] -> [M,96]
  gemm(ymix, out_projw, nullptr, out, M0, C0, E0, E0, E0, C0);
}